// LocallyConnectedXYZLayer_modify_10179072491793
// MI455X (gfx1250) — compile-verified
//
#include <hip/hip_runtime.h>

// LocallyConnectedXYZLayer: bilateral-style 5x5 locally-connected filter.
// xyz:     (8, 3, 64, 2048) f32   (wrap padded)
// softmax: (8,20, 64, 2048) f32   (zero padded, pre-multiplied by mask)
// mask:    (8,    64, 2048) int   (harness passes integers as int32)
// out:     (8,20, 64, 2048) f32

constexpr int NB   = 8;
constexpr int NCH  = 20;
constexpr int HTOT = 64;
constexpr int WTOT = 2048;
constexpr int HW   = HTOT * WTOT;

constexpr int H_TILE = 8;
constexpr int W_TILE = 32;
constexpr int HALO_H = H_TILE + 4;   // 12
constexpr int HALO_W = W_TILE + 4;   // 36
constexpr int NPIX   = HALO_H * HALO_W; // 432
constexpr int BLOCK  = 256;          // 8 wave32s

// ---- CDNA5 async global->LDS copy (gather on global side, scatter on LDS side).
// VDST VGPR = LDS byte address (offset from LDS_BASE), VADDR = 64-bit global addr.
// Tracked by ASYNCcnt; completion enforced with s_wait_asynccnt.
__device__ __forceinline__ void async_g2l_b32(unsigned lds_byte_off, const void* gaddr) {
    asm volatile("global_load_async_to_lds_b32 %0, %1, off"
                 :: "v"(lds_byte_off), "v"(gaddr)
                 : "memory");
}

__device__ __forceinline__ void wait_async_all() {
#if __has_builtin(__builtin_amdgcn_s_wait_asynccnt)
    __builtin_amdgcn_s_wait_asynccnt(0);
#else
    asm volatile("s_wait_asynccnt 0" ::: "memory");
#endif
}

// Low 32 bits of a flat pointer into LDS == byte offset from LDS_BASE
// (ISA 10.2 aperture mapping).
__device__ __forceinline__ unsigned lds_off(const void* p) {
    return (unsigned)(unsigned long long)p;
}

__global__ __launch_bounds__(BLOCK)
void lcxyz_kernel(const float* __restrict__ xyz,
                  const float* __restrict__ softmax,
                  const int*   __restrict__ mask,
                  float*       __restrict__ out) {
    // Pixel-major, channel-contiguous softmax tile: pixel stride = 20 f32 = 80 B (16B aligned).
    __shared__ __align__(16) float s_sm [HALO_H][HALO_W][NCH];
    // xyz interleaved + 4th slot = (mask * in-bounds) factor -> one b128 read per tap.
    __shared__ __align__(16) float s_xyz[HALO_H][HALO_W][4];

    const int n   = blockIdx.z;
    const int th  = blockIdx.y * H_TILE;
    const int tw  = blockIdx.x * W_TILE;
    const int tid = threadIdx.x;

    // ---------------- stage halo tile into LDS (async DMA, layout transform) ----
    for (int i = tid; i < NPIX; i += BLOCK) {
        const int hh = i / HALO_W;
        const int ww = i - hh * HALO_W;
        const int hs = th + hh - 2;              // unpadded coords
        const int ws = tw + ww - 2;
        const int gh = hs & (HTOT - 1);          // wrap (xyz pad mode='wrap')
        const int gw = ws & (WTOT - 1);
        const bool inb = ((unsigned)hs < (unsigned)HTOT) & ((unsigned)ws < (unsigned)WTOT);

        // xyz (wrapped), 3 component planes -> interleaved LDS
        const float* xb = xyz + ((size_t)(n * 3) * HTOT + gh) * WTOT + gw;
        const unsigned lx = lds_off(&s_xyz[hh][ww][0]);
        async_g2l_b32(lx + 0, xb);
        async_g2l_b32(lx + 4, xb + (size_t)HW);
        async_g2l_b32(lx + 8, xb + (size_t)2 * HW);

        // mask * in-bounds factor (zero-pad of softmax folded into the weight)
        float mf = 0.0f;
        if (inb) mf = (mask[((size_t)n * HTOT + gh) * WTOT + gw] != 0) ? 1.0f : 0.0f;
        s_xyz[hh][ww][3] = mf;

        // softmax: 20 channel planes -> channel-contiguous LDS pixel.
        // OOB taps use wrapped (valid) addresses; contribution killed by mf==0.
        const float* sb = softmax + ((size_t)(n * NCH) * HTOT + gh) * WTOT + gw;
        const unsigned ls = lds_off(&s_sm[hh][ww][0]);
        #pragma unroll
        for (int c = 0; c < NCH; ++c)
            async_g2l_b32(ls + 4u * c, sb + (size_t)c * HW);
    }

    wait_async_all();     // this wave's async copies landed in LDS
    __syncthreads();      // all waves' copies visible block-wide

    // ---------------- compute: 25 taps, weight once, 20 FMAs per tap ------------
    const int lh = tid >> 5;        // 0..7
    const int lw = tid & 31;        // 0..31  (lanes = consecutive w -> coalesced)

    const float4 cpix = *(const float4*)&s_xyz[lh + 2][lw + 2][0];
    const float cx = cpix.x, cy = cpix.y, cz = cpix.z;

    float acc[NCH];
    #pragma unroll
    for (int c = 0; c < NCH; ++c) acc[c] = 0.0f;

    #pragma unroll
    for (int di = 0; di < 5; ++di) {
        #pragma unroll
        for (int dj = 0; dj < 5; ++dj) {
            const float4 q = *(const float4*)&s_xyz[lh + di][lw + dj][0];
            const float dx = q.x - cx, dy = q.y - cy, dz = q.z - cz;
            const float d2 = dx * dx + dy * dy + dz * dz;
            // GAUSS_DEN = 2*sigma^2 = 2 ;  w = mask*inb * exp(-d2/2)  (v_exp_f32)
            const float wgt = q.w * __expf(-0.5f * d2);

            const float* sp = &s_sm[lh + di][lw + dj][0];
            #pragma unroll
            for (int c = 0; c < NCH; c += 4) {           // 5x ds_load_b128
                const float4 s = *(const float4*)(sp + c);
                acc[c + 0] += wgt * s.x;
                acc[c + 1] += wgt * s.y;
                acc[c + 2] += wgt * s.z;
                acc[c + 3] += wgt * s.w;
            }
        }
    }

    // ---------------- store (coalesced b32 per channel plane) -------------------
    const int gh = th + lh;
    const int gw = tw + lw;
    float* ob = out + ((size_t)(n * NCH) * HTOT + gh) * WTOT + gw;
    #pragma unroll
    for (int c = 0; c < NCH; ++c)
        ob[(size_t)c * HW] = acc[c];
}

extern "C" void kernel_launch(void* const* d_in, const int* in_sizes, int n_in,
                              void* d_out, int out_size, void* d_ws, size_t ws_size,
                              hipStream_t stream) {
    const float* xyz     = (const float*)d_in[0];
    const float* softmax = (const float*)d_in[1];
    const int*   mask    = (const int*)d_in[2];
    float*       out     = (float*)d_out;

    dim3 grid(WTOT / W_TILE, HTOT / H_TILE, NB);   // 64 x 8 x 8 = 4096 blocks
    lcxyz_kernel<<<grid, BLOCK, 0, stream>>>(xyz, softmax, mask, out);
}